// UnalignmentLoss_68702296867289
// MI455X (gfx1250) — compile-verified
//
#include <hip/hip_runtime.h>

// ---------------------------------------------------------------------------
// Shift-and-compare cost-volume loss for MI455X (gfx1250, wave32).
//   x, y : (4,3,256,256) f32.  For 17x17 even shifts, mean-L1 of shifted x
//   crop vs fixed center crop of y; output = min over shifts.
// Both tensors (6 MB) live in the 192 MB L2 after first touch, so one block
// per shift streaming from L2 is bandwidth-free; reduction uses
// V_WMMA_F32_16X16X4_F32 against an all-ones B matrix for the in-wave sum.
// ---------------------------------------------------------------------------

typedef __attribute__((ext_vector_type(2))) float v2f;
typedef __attribute__((ext_vector_type(8))) float v8f;

constexpr int kTol      = 16;
constexpr int kStride   = 2;
constexpr int kSh       = 17;          // shifts per axis
constexpr int kNShift   = kSh * kSh;   // 289
constexpr int kB        = 4;
constexpr int kC        = 3;
constexpr int kH        = 256;
constexpr int kW        = 256;
constexpr int kCh       = kH - 2 * kTol;   // 224
constexpr int kCw       = kW - 2 * kTol;   // 224
constexpr int kPlaneOut = kCh * kCw;       // 50176
constexpr int kBC       = kB * kC;         // 12
constexpr int kNElem    = kBC * kPlaneOut; // 602112 (== 512 * 1176 exactly)
constexpr int kPlaneIn  = kH * kW;         // 65536
constexpr int kThreads  = 512;             // 16 wave32 waves per block
constexpr int kWaves    = kThreads / 32;

// Wave32 full-sum via WMMA: A holds the per-lane partials (K=0 for lanes
// 0-15, K=2 for lanes 16-31 per the 32-bit 16x4 A layout), B is all-ones.
// D[m][n] = acc[m] + acc[m+16]; summing a lane's 8 D regs gives half the
// wave total; one xor-16 shuffle completes it.  EXEC must be all 1s here.
__device__ __forceinline__ float wave_sum_wmma(float acc) {
  v2f a = {acc, 0.0f};
  v2f b = {1.0f, 1.0f};
  v8f c = {};
  v8f d = __builtin_amdgcn_wmma_f32_16x16x4_f32(
      /*neg_a=*/false, a, /*neg_b=*/false, b,
      /*c_mod=*/(short)0, c, /*reuse_a=*/false, /*reuse_b=*/false);
  float s = d[0] + d[1] + d[2] + d[3] + d[4] + d[5] + d[6] + d[7];
  s += __shfl_xor(s, 16, 32);   // wave32: combine the two half-row sums
  return s;
}

// One block per shift: sum |x[bc, si+r, sj+c] - y[bc, 16+r, 16+c]| over all
// 12 planes of the 224x224 crop.  Uniform trip count -> EXEC stays full.
__global__ void __launch_bounds__(kThreads)
shift_l1_sums(const float* __restrict__ x, const float* __restrict__ y,
              float* __restrict__ sums) {
  const int s   = blockIdx.x;
  const int si  = (s / kSh) * kStride;
  const int sj  = (s % kSh) * kStride;
  const int tid = threadIdx.x;

  float acc = 0.0f;
  for (int e = tid; e < kNElem; e += kThreads) {
    const int bc  = e / kPlaneOut;
    const int rem = e - bc * kPlaneOut;
    const int r   = rem / kCw;
    const int c   = rem - r * kCw;
    const float xv = x[bc * kPlaneIn + (si + r) * kW + (sj + c)];
    const float yv = y[bc * kPlaneIn + (kTol + r) * kW + (kTol + c)];
    acc += __builtin_fabsf(xv - yv);   // v_sub_f32 + v_add_f32 |src|
  }

  const float wsumv = wave_sum_wmma(acc);

  __shared__ float wsum[kWaves];
  const int lane = tid & 31;
  const int wave = tid >> 5;
  if (lane == 0) wsum[wave] = wsumv;
  __syncthreads();

  if (tid == 0) {
    float t = 0.0f;
#pragma unroll
    for (int w = 0; w < kWaves; ++w) t += wsum[w];   // deterministic order
    sums[s] = t;
  }
}

// Min over the 289 per-shift sums, then scale by 1/N for the mean (min and
// the constant positive scale commute).
__global__ void __launch_bounds__(kThreads)
min_mean(const float* __restrict__ sums, float* __restrict__ out) {
  const int tid = threadIdx.x;
  float v = (tid < kNShift) ? sums[tid] : 3.402823466e+38f;
#pragma unroll
  for (int off = 16; off > 0; off >>= 1)
    v = fminf(v, __shfl_xor(v, off, 32));

  __shared__ float wmin[kWaves];
  const int lane = tid & 31;
  const int wave = tid >> 5;
  if (lane == 0) wmin[wave] = v;
  __syncthreads();

  if (tid == 0) {
    float m = wmin[0];
#pragma unroll
    for (int w = 1; w < kWaves; ++w) m = fminf(m, wmin[w]);
    out[0] = m * (1.0f / (float)kNElem);
  }
}

extern "C" void kernel_launch(void* const* d_in, const int* in_sizes, int n_in,
                              void* d_out, int out_size, void* d_ws,
                              size_t ws_size, hipStream_t stream) {
  (void)in_sizes; (void)n_in; (void)out_size; (void)ws_size;
  const float* x = (const float*)d_in[0];
  const float* y = (const float*)d_in[1];
  float* sums = (float*)d_ws;      // 289 f32 partial sums (all overwritten)
  float* out  = (float*)d_out;

  shift_l1_sums<<<dim3(kNShift), dim3(kThreads), 0, stream>>>(x, y, sums);
  min_mean<<<dim3(1), dim3(kThreads), 0, stream>>>(sums, out);
}